// _ImagePool_32581621908016
// MI455X (gfx1250) — compile-verified
//
#include <hip/hip_runtime.h>

#define B_     16
#define CIN    2048
#define COUT   256
#define HW     4096      // 64*64
#define GROUPS 32
#define CPG    (COUT / GROUPS)   // 8 channels per group
#define EPS_GN 1e-5f

typedef __attribute__((ext_vector_type(2))) float v2f;
typedef __attribute__((ext_vector_type(8))) float v8f;

// ---------------------------------------------------------------------------
// Kernel 1: global average pool over HxW.  One 256-thread block per (b,c).
// 512 MB of streaming reads -> dominates runtime; fully coalesced float4.
// ---------------------------------------------------------------------------
__global__ void pool_kernel(const float* __restrict__ x,
                            float* __restrict__ pooled) {
  const int bid = blockIdx.x;                 // b*CIN + c
  const float* p = x + (size_t)bid * HW;
  const int t = threadIdx.x;

  float s = 0.f;
#pragma unroll
  for (int i = 0; i < 4; ++i) {
    float4 v = ((const float4*)p)[t + 256 * i];
    s += v.x + v.y + v.z + v.w;
  }
  // wave32 reduction
#pragma unroll
  for (int off = 16; off > 0; off >>= 1) s += __shfl_down(s, off, 32);

  __shared__ float wsum[8];
  if ((t & 31) == 0) wsum[t >> 5] = s;
  __syncthreads();
  if (t == 0) {
    float tot = 0.f;
#pragma unroll
    for (int w = 0; w < 8; ++w) tot += wsum[w];
    pooled[bid] = tot * (1.0f / (float)HW);
  }
}

// ---------------------------------------------------------------------------
// Kernel 2: D[16x16 tile] = pooled[16x2048] * W^T[2048x16] via
// V_WMMA_F32_16X16X4_F32, then in-block GroupNorm (2 aligned groups of 8
// channels per tile) + affine + ReLU.  One wave (32 threads) per N-tile.
//
// A layout (16x4 f32): lanes 0-15 -> M=lane, VGPR0=K0, VGPR1=K1;
//                      lanes 16-31 -> M=lane-16, VGPR0=K2, VGPR1=K3.
// B layout (4x16 f32) is the transpose-symmetric striping, so both fragments
// are contiguous float2 loads from row-major pooled / W.
// ---------------------------------------------------------------------------
__global__ void gemm_gn_kernel(const float* __restrict__ pooled,
                               const float* __restrict__ W,
                               const float* __restrict__ gamma,
                               const float* __restrict__ beta,
                               float* __restrict__ hn) {
  const int N0   = blockIdx.x * 16;       // channel tile base
  const int lane = threadIdx.x;           // 0..31, full wave (EXEC all 1s)
  const int mn   = lane & 15;             // M for A-frag, N for B-frag
  const int koff = (lane >> 4) * 2;       // lower half: K0/K1, upper: K2/K3

  const float* arow = pooled + (size_t)mn * CIN + koff;
  const float* brow = W + (size_t)(N0 + mn) * CIN + koff;   // B[k][n] = W[n][k]

  v8f acc = {};
#pragma unroll 4
  for (int k = 0; k < CIN; k += 4) {
    v2f a = *(const v2f*)(arow + k);
    v2f b = *(const v2f*)(brow + k);
    // (neg_a, A, neg_b, B, c_mod, C, reuse_a, reuse_b)
    acc = __builtin_amdgcn_wmma_f32_16x16x4_f32(
        false, a, false, b, (short)0, acc, false, false);
  }

  // C/D layout: VGPR i holds (M=i, N=lane) for lanes 0-15 and
  //             (M=i+8, N=lane-16) for lanes 16-31.
  __shared__ float tile[16][17];
  const int rbase = (lane >> 4) * 8;
#pragma unroll
  for (int i = 0; i < 8; ++i) tile[rbase + i][mn] = acc[i];
  __syncthreads();

  // GroupNorm: each row (batch) in this tile holds groups 2n and 2n+1
  // (8 channels each).  32 lanes = 16 rows x 2 groups.
  const int m  = lane >> 1;
  const int c0 = (lane & 1) * CPG;
  float s = 0.f, sq = 0.f;
#pragma unroll
  for (int i = 0; i < CPG; ++i) {
    float v = tile[m][c0 + i];
    s  += v;
    sq += v * v;
  }
  const float mean = s * (1.0f / CPG);
  const float var  = sq * (1.0f / CPG) - mean * mean;
  const float inv  = rsqrtf(var + EPS_GN);
#pragma unroll
  for (int i = 0; i < CPG; ++i) {
    const int ch = N0 + c0 + i;
    float v = (tile[m][c0 + i] - mean) * inv * gamma[ch] + beta[ch];
    hn[m * COUT + ch] = fmaxf(v, 0.f);
  }
}

// ---------------------------------------------------------------------------
// Kernel 3: broadcast hn[b,c] over 64x64 spatial.  One block per (b,c),
// pure coalesced float4 store stream (64 MB).
// ---------------------------------------------------------------------------
__global__ void bcast_kernel(const float* __restrict__ hn,
                             float* __restrict__ out) {
  const int bid = blockIdx.x;             // b*COUT + c
  const float v = hn[bid];
  const float4 v4 = make_float4(v, v, v, v);
  float4* o = (float4*)(out + (size_t)bid * HW);
  const int t = threadIdx.x;
#pragma unroll
  for (int i = 0; i < 4; ++i) o[t + 256 * i] = v4;
}

// ---------------------------------------------------------------------------
extern "C" void kernel_launch(void* const* d_in, const int* in_sizes, int n_in,
                              void* d_out, int out_size, void* d_ws, size_t ws_size,
                              hipStream_t stream) {
  const float* x     = (const float*)d_in[0];   // [16, 2048, 64, 64]
  const float* W     = (const float*)d_in[1];   // [256, 2048]
  const float* gamma = (const float*)d_in[2];   // [256]
  const float* beta  = (const float*)d_in[3];   // [256]
  float* out = (float*)d_out;                   // [16, 256, 64, 64]

  float* pooled = (float*)d_ws;                               // 16*2048 floats
  float* hn     = (float*)((char*)d_ws + (size_t)B_ * CIN * sizeof(float));

  pool_kernel<<<B_ * CIN, 256, 0, stream>>>(x, pooled);
  gemm_gn_kernel<<<COUT / 16, 32, 0, stream>>>(pooled, W, gamma, beta, hn);
  bcast_kernel<<<B_ * COUT, 256, 0, stream>>>(hn, out);
}